// GCN_86921548137090
// MI455X (gfx1250) — compile-verified
//
#include <hip/hip_runtime.h>
#include <hip/hip_bf16.h>
#include <math.h>

typedef __attribute__((ext_vector_type(2))) float v2f;
typedef __attribute__((ext_vector_type(8))) float v8f;

#define KSEL 6

// ---------------------------------------------------------------------------
// Kernel 1: prep. v[c] = fc_w[c] - 1/C ;  av[m,c] = micro_all[m,c]*v[c]
//           S[m] = sum_c micro_all[m,c]^2 v[c] ;  T[n] = sum_c micro[n,c]^2 v[c]
// One block (C threads) per row; rows [0,M) -> micro_all, rows [M,M+N) -> micro.
// ---------------------------------------------------------------------------
__global__ void gcn_prep_kernel(const float* __restrict__ micro,
                                const float* __restrict__ micro_all,
                                const float* __restrict__ fc_w,
                                float* __restrict__ av,
                                float* __restrict__ S,
                                float* __restrict__ T,
                                int N, int M, int C) {
    const int row = blockIdx.x;
    const int c = threadIdx.x;              // blockDim.x == C
    __shared__ float red[256];
    const float v = fc_w[c] - (1.0f / (float)C);
    float val;
    if (row < M) {
        float a = micro_all[row * C + c];
        av[row * C + c] = a * v;
        val = a * a * v;
    } else {
        float b = micro[(row - M) * C + c];
        val = b * b * v;
    }
    red[c] = val;
    __syncthreads();
    for (int s = 128; s > 0; s >>= 1) {
        if (c < s && c + s < C) red[c] += red[c + s];
        __syncthreads();
    }
    if (c == 0) {
        if (row < M) S[row] = red[0];
        else         T[row - M] = red[0];
    }
}

// ---------------------------------------------------------------------------
// Kernel 2: score GEMM via V_WMMA_F32_16X16X4_F32 (full fp32 semantics).
//   anti[n,m] = T[n] + S[m] - 2 * sum_c micro[n,c]*av[m,c]
// One wave per 16x16 tile, 4 waves per block (16 rows x 64 cols per block).
// A-frag layout (ISA 16x4 f32): lanes 0-15 hold K=k,k+1 ; lanes 16-31 K=k+2,k+3
//   -> each lane loads one contiguous float2. B mirrored from row-major av.
// ---------------------------------------------------------------------------
__global__ void gcn_score_kernel(const float* __restrict__ micro, // [N,C]
                                 const float* __restrict__ av,    // [M,C]
                                 const float* __restrict__ S,     // [M]
                                 const float* __restrict__ T,     // [N]
                                 float* __restrict__ anti,        // [N,M]
                                 int N, int M, int C) {
    const int lane = threadIdx.x & 31;
    const int wave = threadIdx.x >> 5;
    const int n0 = blockIdx.y * 16;
    const int m0 = blockIdx.x * 64 + wave * 16;
    const int half = lane >> 4;      // which K-pair this lane owns
    const int l15  = lane & 15;

    const float* arow = micro + (size_t)(n0 + l15) * C;
    const float* brow = av    + (size_t)(m0 + l15) * C;

    v8f acc = {};
    for (int k = 0; k < C; k += 4) {
        const int off = k + 2 * half;
        v2f a = *(const v2f*)(arow + off);
        v2f b = *(const v2f*)(brow + off);
        // D = A(16x4 f32) x B(4x16 f32) + C, fp32 accumulate
        acc = __builtin_amdgcn_wmma_f32_16x16x4_f32(
            /*neg_a=*/false, a, /*neg_b=*/false, b,
            /*c_mod=*/(short)0, acc, /*reuse_a=*/false, /*reuse_b=*/false);
    }

    const float s = S[m0 + l15];
#pragma unroll
    for (int r = 0; r < 8; ++r) {
        const int row = n0 + r + 8 * half;   // C/D layout: VGPR r -> M=r (lanes<16) / r+8
        anti[(size_t)row * M + (m0 + l15)] = T[row] + s - 2.0f * acc[r];
    }
}

// ---------------------------------------------------------------------------
// Kernel 3a: init the loss scalar (stream-ordered before the atomic adds).
// ---------------------------------------------------------------------------
__global__ void gcn_init_loss_kernel(float* loss) { *loss = 1e-4f; }

// ---------------------------------------------------------------------------
// Kernel 3b: fused top-6 softmax + sparse gather + loss. One wave per row n.
// Softmax denominator cancels after mask+renorm, so only top-6 logits matter.
// ---------------------------------------------------------------------------
__global__ void gcn_topk_kernel(const float* __restrict__ anti,      // [N,M]
                                const float* __restrict__ micro,     // [N,C]
                                const float* __restrict__ micro_all, // [M,C]
                                const float* __restrict__ label,     // [N]
                                const float* __restrict__ label_all, // [M]
                                float* __restrict__ out,             // [N*C] + [1]
                                int N, int M, int C) {
    const int row = blockIdx.x;
    const int lane = threadIdx.x;            // blockDim.x == 32 (wave32)
    const float* arow = anti + (size_t)row * M;

    int   sel[KSEL];
    float selv[KSEL];

    // 6 wave-argmax passes over M columns (anti is L2-resident).
    for (int p = 0; p < KSEL; ++p) {
        float best = -INFINITY;
        int   bidx = 0x7fffffff;
        for (int i = 0; i < M / 32; ++i) {
            const int col = i * 32 + lane;
            bool taken = false;
            for (int t = 0; t < p; ++t) taken |= (sel[t] == col);
            const float v = arow[col];
            if (!taken && v > best) { best = v; bidx = col; }
        }
        // wave32 argmax reduction; tie-break on lower index (matches top_k order)
        for (int off = 16; off > 0; off >>= 1) {
            const float ov = __shfl_xor(best, off, 32);
            const int   oi = __shfl_xor(bidx, off, 32);
            if (ov > best || (ov == best && oi < bidx)) { best = ov; bidx = oi; }
        }
        sel[p]  = bidx;
        selv[p] = best;
    }

    // Renormalized softmax weights over the 6 selected logits.
    float mx = selv[0];
    for (int p = 1; p < KSEL; ++p) mx = fmaxf(mx, selv[p]);
    float w[KSEL], wsum = 0.0f;
    for (int p = 0; p < KSEL; ++p) { w[p] = __expf(selv[p] - mx); wsum += w[p]; }
    const float inv = 1.0f / wsum;
    for (int p = 0; p < KSEL; ++p) w[p] *= inv;

    // micro_tmp[row,:] = micro[row,:] + sum_k w_k * micro_all[sel_k,:]
    for (int c = lane; c < C; c += 32) {
        float acc = micro[(size_t)row * C + c];
#pragma unroll
        for (int p = 0; p < KSEL; ++p)
            acc += w[p] * micro_all[(size_t)sel[p] * C + c];
        out[(size_t)row * C + c] = acc;
    }

    // co_loss contribution: sum_k w_k * |label_all[sel_k] - label[row]| / N
    if (lane == 0) {
        const float lb = label[row];
        float contrib = 0.0f;
        for (int p = 0; p < KSEL; ++p)
            contrib += w[p] * fabsf(label_all[sel[p]] - lb);
        atomicAdd(out + (size_t)N * C, contrib / (float)N);
    }
}

// ---------------------------------------------------------------------------
extern "C" void kernel_launch(void* const* d_in, const int* in_sizes, int n_in,
                              void* d_out, int out_size, void* d_ws, size_t ws_size,
                              hipStream_t stream) {
    const float* micro     = (const float*)d_in[0]; // [N,C]
    const float* label     = (const float*)d_in[1]; // [N]
    const float* micro_all = (const float*)d_in[2]; // [M,C]
    const float* label_all = (const float*)d_in[3]; // [M]
    const float* fc_w      = (const float*)d_in[4]; // [C]

    const int N = in_sizes[1];              // 512
    const int C = in_sizes[4];              // 256
    const int M = in_sizes[3];              // 2048

    // Workspace layout (floats): av[M*C] | S[M] | T[N] | anti[N*M]  (~6 MB)
    float* av   = (float*)d_ws;
    float* S    = av + (size_t)M * C;
    float* T    = S + M;
    float* anti = T + N;
    float* out  = (float*)d_out;

    gcn_prep_kernel<<<M + N, C, 0, stream>>>(micro, micro_all, fc_w, av, S, T, N, M, C);

    dim3 sgrid(M / 64, N / 16);
    gcn_score_kernel<<<sgrid, 128, 0, stream>>>(micro, av, S, T, anti, N, M, C);

    gcn_init_loss_kernel<<<1, 1, 0, stream>>>(out + (size_t)N * C);

    gcn_topk_kernel<<<N, 32, 0, stream>>>(anti, micro, micro_all, label, label_all,
                                          out, N, M, C);
}